// GCN_25237227831552
// MI455X (gfx1250) — compile-verified
//
#include <hip/hip_runtime.h>
#include <hip/hip_bf16.h>

#define NODES 50000
#define EDGES 1600000

typedef float v2f __attribute__((ext_vector_type(2)));
typedef float v8f __attribute__((ext_vector_type(8)));

// Hardware float atomic add (global_atomic_add_f32) — avoid CAS-loop fallback.
__device__ __forceinline__ void atom_add_f32(float* p, float v) {
  unsafeAtomicAdd(p, v);
}

// ---------------- utility kernels ----------------

__global__ void gcn_zero_kernel(float* __restrict__ p, long n) {
  long t = (long)blockIdx.x * blockDim.x + threadIdx.x;
  if (t < n) p[t] = 0.0f;
}

__global__ void gcn_deg_count_kernel(const int* __restrict__ rows, float* __restrict__ deg) {
  int e = blockIdx.x * blockDim.x + threadIdx.x;
  if (e < EDGES) atom_add_f32(&deg[rows[e]], 1.0f);  // integer-valued: order-independent, exact
}

__global__ void gcn_dis_kernel(float* __restrict__ deg) {
  int i = blockIdx.x * blockDim.x + threadIdx.x;
  if (i < NODES) {
    float d = deg[i];
    deg[i] = (d > 0.0f) ? (1.0f / sqrtf(d)) : 0.0f;
  }
}

// ---------------- WMMA fp32 GEMM: T[N x FOUT] = X[N x K] @ W[K x FOUT] ----------------
// One block per 16-row slab of X (N/16 = 3125 blocks). blockDim = 32 * (FOUT/16).
// Each wave owns one 16x16 output tile; A-slab staged in LDS (padded stride -> no bank conflicts).
template<int K, int FOUT>
__global__ void gcn_gemm_wmma_kernel(const float* __restrict__ X,
                                     const float* __restrict__ W,
                                     float* __restrict__ T) {
  __shared__ float As[16 * (K + 1)];
  const int rowBase = blockIdx.x * 16;
  for (int i = threadIdx.x; i < 16 * K; i += blockDim.x) {
    int r = i / K, c = i % K;
    As[r * (K + 1) + c] = X[(long)(rowBase + r) * K + c];
  }
  __syncthreads();

  const int wave = threadIdx.x >> 5;
  const int lane = threadIdx.x & 31;
  const int half = lane >> 4;   // 0: lanes 0-15, 1: lanes 16-31
  const int idx  = lane & 15;   // A row / B column within tile
  const int colBase = wave * 16;

  v8f acc = {};
  for (int k0 = 0; k0 < K; k0 += 4) {
    const int ka = k0 + 2 * half;
    v2f a, b;
    // A 16x4 layout: vgpr0 = K=k0 (lanes0-15) / K=k0+2 (lanes16-31); vgpr1 = +1
    a.x = As[idx * (K + 1) + ka];
    a.y = As[idx * (K + 1) + ka + 1];
    // B 4x16 layout: row striped across lanes; vgpr0 = row k0 / k0+2, vgpr1 = +1
    b.x = W[(long)ka * FOUT + colBase + idx];
    b.y = W[(long)(ka + 1) * FOUT + colBase + idx];
    acc = __builtin_amdgcn_wmma_f32_16x16x4_f32(false, a, false, b,
                                                (short)0, acc, false, false);
  }
  // C/D layout: VGPR v holds M = v + 8*half, N = idx
  #pragma unroll
  for (int v = 0; v < 8; ++v)
    T[(long)(rowBase + v + 8 * half) * FOUT + colBase + idx] = acc[v];
}

// ---------------- edge aggregation: agg[row] += dis[row]*dis[col] * src[col] ----------------
// F/4 threads per edge, float4 per thread -> coalesced 512B/256B/16B chunks; L2-resident.
template<int F>
__global__ void gcn_agg_kernel(const float* __restrict__ src,
                               const int* __restrict__ rows,
                               const int* __restrict__ cols,
                               const float* __restrict__ dis,
                               float* __restrict__ agg) {
  constexpr int TPE = F / 4;
  const long t = (long)blockIdx.x * blockDim.x + threadIdx.x;
  const long e = t / TPE;
  const int fg = (int)(t % TPE);
  if (e >= EDGES) return;
  const int r = rows[e];
  const int c = cols[e];
  const float norm = dis[r] * dis[c];
  const float4 xv = *(const float4*)(src + (long)c * F + fg * 4);
  float* dst = agg + (long)r * F + fg * 4;
  atom_add_f32(dst + 0, norm * xv.x);
  atom_add_f32(dst + 1, norm * xv.y);
  atom_add_f32(dst + 2, norm * xv.z);
  atom_add_f32(dst + 3, norm * xv.w);
}

template<int F>
__global__ void gcn_bias_relu_kernel(float* __restrict__ h, const float* __restrict__ b) {
  long t = (long)blockIdx.x * blockDim.x + threadIdx.x;
  if (t < (long)NODES * F) {
    float v = h[t] + b[(int)(t % F)];
    h[t] = v > 0.0f ? v : 0.0f;
  }
}

// ---------------- layer 3 skinny GEMM: T3[N x 4] = H2[N x 64] @ W3[64 x 4] ----------------
__global__ void gcn_gemm_k64_f4_kernel(const float* __restrict__ H,
                                       const float* __restrict__ W,
                                       float* __restrict__ T) {
  int n = blockIdx.x * blockDim.x + threadIdx.x;
  if (n >= NODES) return;
  const float* hr = H + (long)n * 64;
  float a0 = 0.f, a1 = 0.f, a2 = 0.f, a3 = 0.f;
  #pragma unroll 8
  for (int k = 0; k < 64; ++k) {
    float h = hr[k];
    a0 += h * W[k * 4 + 0];
    a1 += h * W[k * 4 + 1];
    a2 += h * W[k * 4 + 2];
    a3 += h * W[k * 4 + 3];
  }
  *(float4*)(T + (long)n * 4) = make_float4(a0, a1, a2, a3);
}

__global__ void gcn_softmax4_kernel(const float* __restrict__ agg,
                                    const float* __restrict__ b,
                                    float* __restrict__ out) {
  int n = blockIdx.x * blockDim.x + threadIdx.x;
  if (n >= NODES) return;
  float v0 = agg[(long)n * 4 + 0] + b[0];
  float v1 = agg[(long)n * 4 + 1] + b[1];
  float v2 = agg[(long)n * 4 + 2] + b[2];
  float v3 = agg[(long)n * 4 + 3] + b[3];
  float m = fmaxf(fmaxf(v0, v1), fmaxf(v2, v3));
  float e0 = expf(v0 - m), e1 = expf(v1 - m), e2 = expf(v2 - m), e3 = expf(v3 - m);
  float inv = 1.0f / (e0 + e1 + e2 + e3);
  *(float4*)(out + (long)n * 4) = make_float4(e0 * inv, e1 * inv, e2 * inv, e3 * inv);
}

// ---------------- launch ----------------

static inline int cdiv(long a, long b) { return (int)((a + b - 1) / b); }

extern "C" void kernel_launch(void* const* d_in, const int* in_sizes, int n_in,
                              void* d_out, int out_size, void* d_ws, size_t ws_size,
                              hipStream_t stream) {
  const float* x  = (const float*)d_in[0];
  const int* rows = (const int*)d_in[1];
  const int* cols = (const int*)d_in[2];
  const float* w1 = (const float*)d_in[3];
  const float* b1 = (const float*)d_in[4];
  const float* w2 = (const float*)d_in[5];
  const float* b2 = (const float*)d_in[6];
  const float* w3 = (const float*)d_in[7];
  const float* b3 = (const float*)d_in[8];
  float* out = (float*)d_out;

  float* ws   = (float*)d_ws;
  float* dis  = ws;                            // N floats
  float* bufA = ws + NODES;                    // N*128 floats
  float* bufB = bufA + (size_t)NODES * 128;    // N*128 floats

  // node degrees -> dis = deg^{-1/2}
  gcn_zero_kernel<<<cdiv(NODES, 256), 256, 0, stream>>>(dis, NODES);
  gcn_deg_count_kernel<<<cdiv(EDGES, 256), 256, 0, stream>>>(rows, dis);
  gcn_dis_kernel<<<cdiv(NODES, 256), 256, 0, stream>>>(dis);

  // Layer 1: T1 = x @ w1 (transform first: A(XW) == (AX)W), then aggregate, bias+relu
  gcn_gemm_wmma_kernel<128, 128><<<NODES / 16, 256, 0, stream>>>(x, w1, bufA);
  gcn_zero_kernel<<<cdiv((long)NODES * 128, 256), 256, 0, stream>>>(bufB, (long)NODES * 128);
  gcn_agg_kernel<128><<<cdiv((long)EDGES * 32, 256), 256, 0, stream>>>(bufA, rows, cols, dis, bufB);
  gcn_bias_relu_kernel<128><<<cdiv((long)NODES * 128, 256), 256, 0, stream>>>(bufB, b1);

  // Layer 2: transform 128->64 BEFORE aggregation (halves scatter/gather traffic)
  gcn_gemm_wmma_kernel<128, 64><<<NODES / 16, 128, 0, stream>>>(bufB, w2, bufA);
  gcn_zero_kernel<<<cdiv((long)NODES * 64, 256), 256, 0, stream>>>(bufB, (long)NODES * 64);
  gcn_agg_kernel<64><<<cdiv((long)EDGES * 16, 256), 256, 0, stream>>>(bufA, rows, cols, dis, bufB);
  gcn_bias_relu_kernel<64><<<cdiv((long)NODES * 64, 256), 256, 0, stream>>>(bufB, b2);

  // Layer 3: transform 64->4 BEFORE aggregation (16x less traffic), then softmax
  gcn_gemm_k64_f4_kernel<<<cdiv(NODES, 256), 256, 0, stream>>>(bufB, w3, bufA);
  float* agg3 = bufA + (size_t)NODES * 4;
  gcn_zero_kernel<<<cdiv((long)NODES * 4, 256), 256, 0, stream>>>(agg3, (long)NODES * 4);
  gcn_agg_kernel<4><<<cdiv((long)EDGES, 256), 256, 0, stream>>>(bufA, rows, cols, dis, agg3);
  gcn_softmax4_kernel<<<cdiv(NODES, 256), 256, 0, stream>>>(agg3, b3, out);
}